// AdapterSubnet_59330678227175
// MI455X (gfx1250) — compile-verified
//
#include <hip/hip_runtime.h>
#include <hip/hip_bf16.h>
#include <stdint.h>

#define NBINS 4096
#define HIST_BLOCK 256
#define TILE_ELEMS 1024   // 256 threads * 4 floats (b128 per lane)

typedef float v4f __attribute__((ext_vector_type(4)));
typedef int v4i_gcc __attribute__((vector_size(16)));   // matches builtin param pointee

// ---- CDNA5 async global->LDS path (guarded; fallback keeps it compiling) ----
#if defined(__HIP_DEVICE_COMPILE__) && __has_builtin(__builtin_amdgcn_global_load_async_to_lds_b128)
#define USE_ASYNC_BUILTIN 1
#else
#define USE_ASYNC_BUILTIN 0
#endif

#if defined(__HIP_DEVICE_COMPILE__) && __has_builtin(__builtin_amdgcn_s_wait_asynccnt)
#define USE_WAITASYNC_BUILTIN 1
#else
#define USE_WAITASYNC_BUILTIN 0
#endif

__device__ __forceinline__ void async_copy_b128(const void* g, void* l) {
#if USE_ASYNC_BUILTIN
  __builtin_amdgcn_global_load_async_to_lds_b128(
      (__attribute__((address_space(1))) v4i_gcc*)g,
      (__attribute__((address_space(3))) v4i_gcc*)l,
      0, 0);
#elif defined(__HIP_DEVICE_COMPILE__)
  unsigned long long ga = (unsigned long long)g;
  unsigned la = (unsigned)(unsigned long long)(__attribute__((address_space(3))) char*)l;
  asm volatile("global_load_async_to_lds_b128 %0, %1, off" :: "v"(la), "v"(ga) : "memory");
#else
  *(uint4*)l = *(const uint4*)g;  // host-pass parse only
#endif
}

__device__ __forceinline__ void wait_async_le1() {
#if USE_WAITASYNC_BUILTIN
  __builtin_amdgcn_s_wait_asynccnt(1);
#elif defined(__HIP_DEVICE_COMPILE__)
  asm volatile("s_wait_asynccnt 1" ::: "memory");
#endif
  asm volatile("" ::: "memory");
}

__device__ __forceinline__ void wait_async_0() {
#if USE_WAITASYNC_BUILTIN
  __builtin_amdgcn_s_wait_asynccnt(0);
#elif defined(__HIP_DEVICE_COMPILE__)
  asm volatile("s_wait_asynccnt 0" ::: "memory");
#endif
  asm volatile("" ::: "memory");
}

// ---------------------------------------------------------------------------

__global__ void zero_ws_kernel(unsigned* __restrict__ ws, int n) {
  int i = blockIdx.x * blockDim.x + threadIdx.x;
  if (i < n) ws[i] = 0u;
}

// Pass 1: histogram of top 12 bits of |x| bit pattern. Async double-buffered LDS staging.
__global__ void hist_pass1(const float* __restrict__ x, unsigned n,
                           unsigned* __restrict__ ghist) {
  __shared__ unsigned shist[NBINS];
  __shared__ uint4 stage[2][HIST_BLOCK];
  const int tid = threadIdx.x;

  for (int i = tid; i < NBINS; i += blockDim.x) shist[i] = 0u;
  __syncthreads();

  const unsigned tiles = n / TILE_ELEMS;
  const unsigned tstride = gridDim.x;
  unsigned tile = blockIdx.x;
  int buf = 0;

  if (tile < tiles)
    async_copy_b128(x + (size_t)tile * TILE_ELEMS + (size_t)tid * 4, &stage[0][tid]);

  while (tile < tiles) {
    const unsigned next = tile + tstride;
    if (next < tiles) {
      async_copy_b128(x + (size_t)next * TILE_ELEMS + (size_t)tid * 4, &stage[buf ^ 1][tid]);
      wait_async_le1();   // oldest (current buf) complete; asyncs retire in order
    } else {
      wait_async_0();
    }
    const uint4 v = stage[buf][tid];
    atomicAdd(&shist[(v.x & 0x7fffffffu) >> 20], 1u);
    atomicAdd(&shist[(v.y & 0x7fffffffu) >> 20], 1u);
    atomicAdd(&shist[(v.z & 0x7fffffffu) >> 20], 1u);
    atomicAdd(&shist[(v.w & 0x7fffffffu) >> 20], 1u);
    buf ^= 1;
    tile = next;
  }

  // scalar tail (n not multiple of TILE_ELEMS)
  for (unsigned i = tiles * TILE_ELEMS + blockIdx.x * blockDim.x + threadIdx.x;
       i < n; i += gridDim.x * blockDim.x) {
    atomicAdd(&shist[(__float_as_uint(x[i]) & 0x7fffffffu) >> 20], 1u);
  }

  __syncthreads();
  for (int i = tid; i < NBINS; i += blockDim.x) {
    const unsigned c = shist[i];
    if (c) atomicAdd(&ghist[i], c);
  }
}

// Pass 2: histogram of bits[19:8] for elements whose top-12 bin == b1 (ctrl[0]).
__global__ void hist_pass2(const float* __restrict__ x, unsigned n,
                           const unsigned* __restrict__ ctrl,
                           unsigned* __restrict__ ghist2) {
  __shared__ unsigned shist[NBINS];
  __shared__ uint4 stage[2][HIST_BLOCK];
  const int tid = threadIdx.x;
  const unsigned b1 = ctrl[0];

  for (int i = tid; i < NBINS; i += blockDim.x) shist[i] = 0u;
  __syncthreads();

  const unsigned tiles = n / TILE_ELEMS;
  const unsigned tstride = gridDim.x;
  unsigned tile = blockIdx.x;
  int buf = 0;

  if (tile < tiles)
    async_copy_b128(x + (size_t)tile * TILE_ELEMS + (size_t)tid * 4, &stage[0][tid]);

  while (tile < tiles) {
    const unsigned next = tile + tstride;
    if (next < tiles) {
      async_copy_b128(x + (size_t)next * TILE_ELEMS + (size_t)tid * 4, &stage[buf ^ 1][tid]);
      wait_async_le1();
    } else {
      wait_async_0();
    }
    const uint4 v = stage[buf][tid];
    unsigned k;
    k = v.x & 0x7fffffffu; if ((k >> 20) == b1) atomicAdd(&shist[(k >> 8) & 0xfffu], 1u);
    k = v.y & 0x7fffffffu; if ((k >> 20) == b1) atomicAdd(&shist[(k >> 8) & 0xfffu], 1u);
    k = v.z & 0x7fffffffu; if ((k >> 20) == b1) atomicAdd(&shist[(k >> 8) & 0xfffu], 1u);
    k = v.w & 0x7fffffffu; if ((k >> 20) == b1) atomicAdd(&shist[(k >> 8) & 0xfffu], 1u);
    buf ^= 1;
    tile = next;
  }

  for (unsigned i = tiles * TILE_ELEMS + blockIdx.x * blockDim.x + threadIdx.x;
       i < n; i += gridDim.x * blockDim.x) {
    const unsigned k = __float_as_uint(x[i]) & 0x7fffffffu;
    if ((k >> 20) == b1) atomicAdd(&shist[(k >> 8) & 0xfffu], 1u);
  }

  __syncthreads();
  for (int i = tid; i < NBINS; i += blockDim.x) {
    const unsigned c = shist[i];
    if (c) atomicAdd(&ghist2[i], c);
  }
}

// Single-lane suffix scan of a 4096-bin histogram.
// level 1: ctrl[0]=b1, ctrl[1]=r1   level 2: ctrl[2]=thresh24, ctrl[3]=r2, ctrl[4]=0
__global__ void select_kernel(const unsigned* __restrict__ hist,
                              unsigned* __restrict__ ctrl,
                              unsigned K, int level) {
  if (threadIdx.x != 0 || blockIdx.x != 0) return;
  const unsigned want = (level == 1) ? K : ctrl[1];
  unsigned long long cum = 0;
  unsigned cnt = 0;
  int b = 0;
  for (int i = NBINS - 1; i >= 0; --i) {
    cnt = hist[i];
    cum += cnt;
    if (cum >= (unsigned long long)want) { b = i; break; }
  }
  const unsigned r = want - (unsigned)(cum - cnt);
  if (level == 1) {
    ctrl[0] = (unsigned)b;
    ctrl[1] = r;
  } else {
    ctrl[2] = (ctrl[0] << 12) | (unsigned)b;  // 24-bit prefix threshold
    ctrl[3] = r;
    ctrl[4] = 0u;                             // tie counter
  }
}

__device__ __forceinline__ float classify_one(unsigned bits, unsigned thresh,
                                              unsigned r2, unsigned* tie) {
  const unsigned p = (bits & 0x7fffffffu) >> 8;
  if (p > thresh) return 1.0f;
  if (p < thresh) return 0.0f;
  const unsigned old = atomicAdd(tie, 1u);
  return (old < r2) ? 1.0f : 0.0f;
}

// Final streaming pass: b128 loads, NT b128 stores, prefetch ahead.
__global__ void write_mask(const float* __restrict__ x, float* __restrict__ out,
                           unsigned n, const unsigned* __restrict__ ctrl,
                           unsigned* __restrict__ tie) {
  const unsigned thresh = ctrl[2];
  const unsigned r2 = ctrl[3];
  const unsigned nv = n >> 2;
  const unsigned gstride = gridDim.x * blockDim.x;
  for (unsigned i = blockIdx.x * blockDim.x + threadIdx.x; i < nv; i += gstride) {
    __builtin_prefetch(x + (size_t)(i + gstride) * 4, 0, 0);  // global_prefetch_b8
    const uint4 v = *(const uint4*)(x + (size_t)i * 4);
    v4f o;
    o.x = classify_one(v.x, thresh, r2, tie);
    o.y = classify_one(v.y, thresh, r2, tie);
    o.z = classify_one(v.z, thresh, r2, tie);
    o.w = classify_one(v.w, thresh, r2, tie);
    __builtin_nontemporal_store(o, (v4f*)(out + (size_t)i * 4));
  }
  for (unsigned i = (nv << 2) + blockIdx.x * blockDim.x + threadIdx.x; i < n; i += gstride) {
    out[i] = classify_one(__float_as_uint(x[i]), thresh, r2, tie);
  }
}

extern "C" void kernel_launch(void* const* d_in, const int* in_sizes, int n_in,
                              void* d_out, int out_size, void* d_ws, size_t ws_size,
                              hipStream_t stream) {
  (void)n_in; (void)out_size; (void)ws_size;
  const float* down = (const float*)d_in[0];
  const float* up   = (const float*)d_in[1];
  float* out = (float*)d_out;
  const unsigned n0 = (unsigned)in_sizes[0];
  const unsigned n1 = (unsigned)in_sizes[1];

  unsigned* ws    = (unsigned*)d_ws;
  unsigned* hist1 = ws;                 // NBINS u32
  unsigned* hist2 = ws + NBINS;         // NBINS u32
  unsigned* ctrl  = ws + 2 * NBINS;     // 16 u32 control block

  auto run_one = [&](const float* x, float* o, unsigned n) {
    if (n == 0) return;
    // Match Python: j = int((1.0 - 0.1) * n); keep = n - j
    const long long j = (long long)((1.0 - 0.1) * (double)n);
    const unsigned K = (unsigned)((long long)n - j);

    const int zn = 2 * NBINS + 16;
    zero_ws_kernel<<<(zn + 255) / 256, 256, 0, stream>>>(ws, zn);

    unsigned tiles = n / TILE_ELEMS;
    int hblocks = 2048;
    if (tiles > 0 && (unsigned)hblocks > tiles) hblocks = (int)tiles;
    if (hblocks < 1) hblocks = 1;

    hist_pass1<<<hblocks, HIST_BLOCK, 0, stream>>>(x, n, hist1);
    select_kernel<<<1, 32, 0, stream>>>(hist1, ctrl, K, 1);
    hist_pass2<<<hblocks, HIST_BLOCK, 0, stream>>>(x, n, ctrl, hist2);
    select_kernel<<<1, 32, 0, stream>>>(hist2, ctrl, K, 2);

    const int wblocks = 4096;
    write_mask<<<wblocks, 256, 0, stream>>>(x, o, n, ctrl, ctrl + 4);
  };

  run_one(down, out, n0);
  run_one(up, out + n0, n1);
}